// AutoEncoderLSTM_63556926046784
// MI455X (gfx1250) — compile-verified
//
#include <hip/hip_runtime.h>
#include <hip/hip_bf16.h>

// Sizes from the reference
#define LSEQ 4096
#define TDIM 25
#define BDIM 257
#define NROWS (LSEQ * BDIM)          // 1,052,672 = 65,792 * 16
#define NTILES (NROWS / 16)          // 65,792
#define WAVES_PER_BLOCK 8
#define NBLOCKS (NTILES / WAVES_PER_BLOCK)   // 8,224

typedef float v2f __attribute__((ext_vector_type(2)));
typedef float v8f __attribute__((ext_vector_type(8)));

__device__ __forceinline__ float elu_f(float v) {
    return v > 0.0f ? v : (__expf(v) - 1.0f);
}
__device__ __forceinline__ float sigm_f(float v) {
    return 1.0f / (1.0f + __expf(-v));
}

// ---------------------------------------------------------------------------
// Encoder: z0[l,b,0:2] = elu(elu(elu(x[l,b,:]·W_enc^T+b)·W_enc2^T+b2)·W_enc3^T+b3)
// Layer 1 (25->8) via V_WMMA_F32_16X16X4_F32 (7 K-chunks, all fragments
// preloaded branch-free), layers 2/3 per-row VALU via LDS.
// ---------------------------------------------------------------------------
__global__ __launch_bounds__(256) void enc_kernel(
    const float* __restrict__ x,
    const float* __restrict__ W_enc,  const float* __restrict__ b_enc,
    const float* __restrict__ W_enc2, const float* __restrict__ b_enc2,
    const float* __restrict__ W_enc3, const float* __restrict__ b_enc3,
    float* __restrict__ z0)
{
    const int lane = threadIdx.x & 31;
    const int w    = threadIdx.x >> 5;
    const int m    = lane & 15;        // row-in-tile (A) / col n (B,D)
    const int hi   = lane >> 4;        // lane-group selects K pair
    const int tile = blockIdx.x * WAVES_PER_BLOCK + w;
    const int row  = tile * 16 + m;    // row = l*257 + b
    const int l    = row / BDIM;
    const int b    = row - l * BDIM;
    const int xbase = (l * TDIM) * BDIM + b;   // + t*BDIM

    // ---- Preload ALL fragments branch-free (clamped address + mask) ----
    const int mrow = (m < 8) ? m : 0;          // clamped W_enc row
    v2f afrag[7], bfrag[7];
#pragma unroll
    for (int chunk = 0; chunk < 7; ++chunk) {
        const int kb = chunk * 4;
#pragma unroll
        for (int j = 0; j < 2; ++j) {
            const int k  = kb + 2 * hi + j;
            const int kc = (k < TDIM) ? k : 0; // clamp (only chunk 6 can exceed)
            const bool kok = (k < TDIM);
            float av = x[xbase + kc * BDIM];
            float bv = W_enc[mrow * TDIM + kc];
            afrag[chunk][j] = kok ? av : 0.0f;
            bfrag[chunk][j] = (kok && m < 8) ? bv : 0.0f;
        }
    }

    // C init: bias depends only on column n==m
    const float bias_n = (m < 8) ? b_enc[m] : 0.0f;
    v8f acc;
#pragma unroll
    for (int r = 0; r < 8; ++r) acc[r] = bias_n;

    // ---- 7 back-to-back fp32 WMMAs ----
#pragma unroll
    for (int chunk = 0; chunk < 7; ++chunk) {
        acc = __builtin_amdgcn_wmma_f32_16x16x4_f32(
            false, afrag[chunk], false, bfrag[chunk], (short)0, acc, false, false);
    }

    // elu(layer1) -> LDS tile [row][col<8]
    __shared__ float t1[WAVES_PER_BLOCK][16][8];
    if (m < 8) {
#pragma unroll
        for (int r = 0; r < 8; ++r)
            t1[w][r + 8 * hi][m] = elu_f(acc[r]);
    }
    __syncthreads();

    // layers 2 (8->4) and 3 (4->2): one row per lane (lanes 0..15)
    if (hi == 0) {
        float zr[8];
#pragma unroll
        for (int k = 0; k < 8; ++k) zr[k] = t1[w][m][k];
        float d2[4];
#pragma unroll
        for (int n = 0; n < 4; ++n) {
            float s = b_enc2[n];
#pragma unroll
            for (int k = 0; k < 8; ++k) s += W_enc2[n * 8 + k] * zr[k];
            d2[n] = elu_f(s);
        }
        float d3[2];
#pragma unroll
        for (int n = 0; n < 2; ++n) {
            float s = b_enc3[n];
#pragma unroll
            for (int k = 0; k < 4; ++k) s += W_enc3[n * 4 + k] * d2[k];
            d3[n] = elu_f(s);
        }
        z0[row * 2 + 0] = d3[0];
        z0[row * 2 + 1] = d3[1];
    }
}

// ---------------------------------------------------------------------------
// Fused sequential middle: LSTM1 -> elu -> knob-mix -> elu -> LSTM2 -> elu.
// One thread per batch lane b (257 lanes), sequential over L=4096 steps.
// ---------------------------------------------------------------------------
__global__ void lstm_kernel(
    const float* __restrict__ z0, const float* __restrict__ knobs,
    const float* __restrict__ Wih1, const float* __restrict__ Whh1,
    const float* __restrict__ bih1, const float* __restrict__ bhh1,
    const float* __restrict__ W_knobs, const float* __restrict__ b_knobs,
    const float* __restrict__ Wih2, const float* __restrict__ Whh2,
    const float* __restrict__ bih2, const float* __restrict__ bhh2,
    float* __restrict__ z2)
{
    const int b = blockIdx.x * blockDim.x + threadIdx.x;
    if (b >= BDIM) return;

    float wi1[8][2], wh1[8][2], bb1[8];
    float wi2[8][2], wh2[8][2], bb2[8];
#pragma unroll
    for (int r = 0; r < 8; ++r) {
        wi1[r][0] = Wih1[r * 2 + 0]; wi1[r][1] = Wih1[r * 2 + 1];
        wh1[r][0] = Whh1[r * 2 + 0]; wh1[r][1] = Whh1[r * 2 + 1];
        bb1[r] = bih1[r] + bhh1[r];
        wi2[r][0] = Wih2[r * 2 + 0]; wi2[r][1] = Wih2[r * 2 + 1];
        wh2[r][0] = Whh2[r * 2 + 0]; wh2[r][1] = Whh2[r * 2 + 1];
        bb2[r] = bih2[r] + bhh2[r];
    }
    float wk[2][5], bk[2];
#pragma unroll
    for (int j = 0; j < 2; ++j) {
#pragma unroll
        for (int u = 0; u < 5; ++u) wk[j][u] = W_knobs[j * 5 + u];
        bk[j] = b_knobs[j];
    }

    float h1[2] = {0.f, 0.f}, c1[2] = {0.f, 0.f};
    float h2[2] = {0.f, 0.f}, c2[2] = {0.f, 0.f};

    for (int t = 0; t < LSEQ; ++t) {
        const int base = (t * BDIM + b) * 2;
        float zi[2] = {z0[base], z0[base + 1]};
        const float kn0 = knobs[t * 3 + 0];
        const float kn1 = knobs[t * 3 + 1];
        const float kn2 = knobs[t * 3 + 2];

        // LSTM1 cell (gate order i,f,g,o)
        float g[8];
#pragma unroll
        for (int r = 0; r < 8; ++r)
            g[r] = bb1[r] + wi1[r][0] * zi[0] + wi1[r][1] * zi[1]
                          + wh1[r][0] * h1[0] + wh1[r][1] * h1[1];
#pragma unroll
        for (int j = 0; j < 2; ++j) {
            float ii = sigm_f(g[j]),  ff = sigm_f(g[2 + j]);
            float gg = tanhf(g[4 + j]), oo = sigm_f(g[6 + j]);
            c1[j] = ff * c1[j] + ii * gg;
            h1[j] = oo * tanhf(c1[j]);
        }
        float ze0 = elu_f(h1[0]), ze1 = elu_f(h1[1]);

        // knob mix (concat [z, knobs] @ W_knobs^T), elu
        float zk[2];
#pragma unroll
        for (int j = 0; j < 2; ++j) {
            float s = bk[j] + wk[j][0] * ze0 + wk[j][1] * ze1
                            + wk[j][2] * kn0 + wk[j][3] * kn1 + wk[j][4] * kn2;
            zk[j] = elu_f(s);
        }

        // LSTM2 cell
#pragma unroll
        for (int r = 0; r < 8; ++r)
            g[r] = bb2[r] + wi2[r][0] * zk[0] + wi2[r][1] * zk[1]
                          + wh2[r][0] * h2[0] + wh2[r][1] * h2[1];
#pragma unroll
        for (int j = 0; j < 2; ++j) {
            float ii = sigm_f(g[j]),  ff = sigm_f(g[2 + j]);
            float gg = tanhf(g[4 + j]), oo = sigm_f(g[6 + j]);
            c2[j] = ff * c2[j] + ii * gg;
            h2[j] = oo * tanhf(c2[j]);
        }
        z2[base]     = elu_f(h2[0]);
        z2[base + 1] = elu_f(h2[1]);
    }
}

// ---------------------------------------------------------------------------
// Decoder: d3=elu(2->4), d2=elu(4->8) per row; 8->25 via WMMA f32 16x16x4
// (two N-tiles, fragments preloaded branch-free); residual + b_dec + elu;
// coalesced stores via LDS transpose.
// ---------------------------------------------------------------------------
__global__ __launch_bounds__(256) void dec_kernel(
    const float* __restrict__ z2, const float* __restrict__ x,
    const float* __restrict__ W_dec3, const float* __restrict__ b_dec3,
    const float* __restrict__ W_dec2, const float* __restrict__ b_dec2,
    const float* __restrict__ W_dec,  const float* __restrict__ b_dec,
    float* __restrict__ out)
{
    const int lane = threadIdx.x & 31;
    const int w    = threadIdx.x >> 5;
    const int m    = lane & 15;
    const int hi   = lane >> 4;
    const int tile = blockIdx.x * WAVES_PER_BLOCK + w;
    const int row  = tile * 16 + m;    // row = l*257 + b
    const int l    = row / BDIM;
    const int b    = row - l * BDIM;
    const int xbase = (l * TDIM) * BDIM + b;

    __shared__ float a_lds[WAVES_PER_BLOCK][16][8];
    __shared__ float o_lds[WAVES_PER_BLOCK][16][26];

    // ---- Preload B fragments for both N-tiles, branch-free ----
    v2f bfrag[2][2];   // [ntile][chunk]
#pragma unroll
    for (int nt = 0; nt < 2; ++nt) {
        const int n  = nt * 16 + m;
        const int nc = (n < TDIM) ? n : 0;
        const bool nok = (n < TDIM);
#pragma unroll
        for (int chunk = 0; chunk < 2; ++chunk) {
#pragma unroll
            for (int j = 0; j < 2; ++j) {
                const int k = chunk * 4 + 2 * hi + j;
                float v = W_dec[nc * 8 + k];
                bfrag[nt][chunk][j] = nok ? v : 0.0f;
            }
        }
    }

    // Stage 1: per-row 2->4->8 (lanes 0..15)
    if (hi == 0) {
        float zi[2] = {z2[row * 2 + 0], z2[row * 2 + 1]};
        float d3[4];
#pragma unroll
        for (int n = 0; n < 4; ++n) {
            float s = b_dec3[n] + W_dec3[n * 2 + 0] * zi[0] + W_dec3[n * 2 + 1] * zi[1];
            d3[n] = elu_f(s);
        }
#pragma unroll
        for (int n = 0; n < 8; ++n) {
            float s = b_dec2[n];
#pragma unroll
            for (int k = 0; k < 4; ++k) s += W_dec2[n * 4 + k] * d3[k];
            a_lds[w][m][n] = elu_f(s);
        }
    }
    __syncthreads();

    // Stage 2: 8->25 with WMMA; A fragments for K-chunks kb=0,4
    v2f afrag[2];
#pragma unroll
    for (int j = 0; j < 2; ++j) {
        afrag[0][j] = a_lds[w][m][0 + 2 * hi + j];
        afrag[1][j] = a_lds[w][m][4 + 2 * hi + j];
    }
#pragma unroll
    for (int nt = 0; nt < 2; ++nt) {
        const int n = nt * 16 + m;      // output column (t index), valid < 25
        v8f acc;
#pragma unroll
        for (int r = 0; r < 8; ++r) acc[r] = 0.0f;
#pragma unroll
        for (int chunk = 0; chunk < 2; ++chunk) {
            acc = __builtin_amdgcn_wmma_f32_16x16x4_f32(
                false, afrag[chunk], false, bfrag[nt][chunk], (short)0, acc, false, false);
        }
        if (n < TDIM) {
#pragma unroll
            for (int r = 0; r < 8; ++r)
                o_lds[w][r + 8 * hi][n] = acc[r];
        }
    }
    __syncthreads();

    // Stage 3: residual + bias + elu, coalesced across b (lanes 0..15 = rows)
    for (int tt = hi; tt < TDIM; tt += 2) {
        const int idx = xbase + tt * BDIM;
        float v = o_lds[w][m][tt] + b_dec[tt] + x[idx];
        out[idx] = elu_f(v);
    }
}

extern "C" void kernel_launch(void* const* d_in, const int* in_sizes, int n_in,
                              void* d_out, int out_size, void* d_ws, size_t ws_size,
                              hipStream_t stream) {
    const float* x       = (const float*)d_in[0];
    const float* knobs   = (const float*)d_in[1];
    const float* W_enc   = (const float*)d_in[2];
    const float* b_enc   = (const float*)d_in[3];
    const float* W_enc2  = (const float*)d_in[4];
    const float* b_enc2  = (const float*)d_in[5];
    const float* W_enc3  = (const float*)d_in[6];
    const float* b_enc3  = (const float*)d_in[7];
    const float* Wih1    = (const float*)d_in[8];
    const float* Whh1    = (const float*)d_in[9];
    const float* bih1    = (const float*)d_in[10];
    const float* bhh1    = (const float*)d_in[11];
    const float* W_knobs = (const float*)d_in[12];
    const float* b_knobs = (const float*)d_in[13];
    const float* Wih2    = (const float*)d_in[14];
    const float* Whh2    = (const float*)d_in[15];
    const float* bih2    = (const float*)d_in[16];
    const float* bhh2    = (const float*)d_in[17];
    const float* W_dec3  = (const float*)d_in[18];
    const float* b_dec3  = (const float*)d_in[19];
    const float* W_dec2  = (const float*)d_in[20];
    const float* b_dec2  = (const float*)d_in[21];
    const float* W_dec   = (const float*)d_in[22];
    const float* b_dec   = (const float*)d_in[23];
    float* out = (float*)d_out;

    float* z0 = (float*)d_ws;                       // (L, 257, 2) fp32
    float* z2 = z0 + (size_t)NROWS * 2;             // (L, 257, 2) fp32

    enc_kernel<<<NBLOCKS, 256, 0, stream>>>(x, W_enc, b_enc, W_enc2, b_enc2,
                                            W_enc3, b_enc3, z0);
    lstm_kernel<<<1, 288, 0, stream>>>(z0, knobs, Wih1, Whh1, bih1, bhh1,
                                       W_knobs, b_knobs, Wih2, Whh2, bih2, bhh2, z2);
    dec_kernel<<<NBLOCKS, 256, 0, stream>>>(z2, x, W_dec3, b_dec3, W_dec2, b_dec2,
                                            W_dec, b_dec, out);
}